// LinearPPI_64768106824289
// MI455X (gfx1250) — compile-verified
//
#include <hip/hip_runtime.h>

typedef __attribute__((ext_vector_type(2))) float v2f;
typedef __attribute__((ext_vector_type(8))) float v8f;

#define G_GROUPS 4000
#define BLK 8
#define DEG 32
#define BATCH 128
#define DIM (G_GROUPS * BLK)                  // 32000
#define NBLOCKS (G_GROUPS * DEG + G_GROUPS)   // 132000
#define SCAN_T 1024

// Edge descriptor: everything the gather loop needs, resolved at CSR-build time
// so the hot loop does a single s_load_b64 per edge (no dependent indirection).
struct EdgeDesc {
    int woff;   // float offset of w[n]           = n * 64
    int xoff;   // float offset of x column block = block_in[n] * 8
};

// ---------------- CSR build: group edges by block_out ----------------

__global__ void init_kernel(int* __restrict__ count,
                            EdgeDesc* __restrict__ desc) {
    int i = blockIdx.x * blockDim.x + threadIdx.x;
    if (i < G_GROUPS) count[i] = 0;
    // Two sentinel descriptors past the end: the gather loop's unconditional
    // 2-ahead lookahead may load (never consume) them. Point at valid memory.
    if (i < 2) {
        EdgeDesc z; z.woff = 0; z.xoff = 0;
        desc[NBLOCKS + i] = z;
    }
}

__global__ void count_kernel(const int* __restrict__ block_out,
                             int* __restrict__ count) {
    int n = blockIdx.x * blockDim.x + threadIdx.x;
    if (n < NBLOCKS) atomicAdd(&count[block_out[n]], 1);
}

// Single-workgroup exclusive scan of count[0..G) -> offsets[0..G], cursor copy.
__global__ __launch_bounds__(SCAN_T) void scan_kernel(
    const int* __restrict__ count,
    int* __restrict__ offsets,
    int* __restrict__ cursor)
{
    __shared__ int buf[SCAN_T];
    __shared__ int carry_s;
    const int t = threadIdx.x;
    if (t == 0) carry_s = 0;
    __syncthreads();
    for (int base = 0; base < G_GROUPS; base += SCAN_T) {
        const int i = base + t;
        const int v = (i < G_GROUPS) ? count[i] : 0;
        buf[t] = v;
        __syncthreads();
        for (int off = 1; off < SCAN_T; off <<= 1) {   // Hillis-Steele inclusive
            int add = (t >= off) ? buf[t - off] : 0;
            __syncthreads();
            buf[t] += add;
            __syncthreads();
        }
        const int cbase = carry_s;
        const int excl  = cbase + buf[t] - v;
        if (i < G_GROUPS) { offsets[i] = excl; cursor[i] = excl; }
        __syncthreads();
        if (t == 0) carry_s = cbase + buf[SCAN_T - 1];
        __syncthreads();
    }
    if (t == 0) offsets[G_GROUPS] = carry_s;
}

__global__ void fill_kernel(const int* __restrict__ block_out,
                            const int* __restrict__ block_in,
                            int* __restrict__ cursor,
                            EdgeDesc* __restrict__ desc) {
    int n = blockIdx.x * blockDim.x + threadIdx.x;
    if (n < NBLOCKS) {
        int pos = atomicAdd(&cursor[block_out[n]], 1);
        EdgeDesc d;
        d.woff = n * (BLK * BLK);          // pre-resolved w offset
        d.xoff = block_in[n] * BLK;        // pre-resolved x offset
        desc[pos] = d;
    }
}

// ---------------- Gather GEMM: one wave per (group, batch-tile) ----------------
//
// out[:, g, :] = x[:, g, :] + sum_{e : block_out[e]==g}  x[:, block_in[e], :] @ w[e]
// 16x8 @ 8x8 per edge per tile = two chained V_WMMA_F32_16X16X4_F32, C held in
// registers across all ~33 edges of the group -> zero float atomics.

__device__ __forceinline__ void edge_accum(
    const float* __restrict__ x, const float* __restrict__ w,
    EdgeDesc d, int rowbase, int h, int col, float mask, v8f& C)
{
    // Descriptor fields are wave-uniform -> force SGPRs (scalar base addressing).
    const int woff = __builtin_amdgcn_readfirstlane(d.woff);
    const int xoff = __builtin_amdgcn_readfirstlane(d.xoff);

    // B operands (w[n] is 8x8 row-major). All lanes load valid addresses
    // (lanes m>=8 duplicate lanes m-8), then mask -> no EXEC change before WMMA.
    const float* wn = w + woff;
    v2f B1, B2;
    B1.x = wn[(2 * h + 0) * BLK + col] * mask;   // K = 2h
    B1.y = wn[(2 * h + 1) * BLK + col] * mask;   // K = 2h+1
    B2.x = wn[(2 * h + 4) * BLK + col] * mask;   // K = 4+2h
    B2.y = wn[(2 * h + 5) * BLK + col] * mask;   // K = 5+2h

    // A operands: x row 'rowbase', cols {2h,2h+1} and {4+2h,5+2h} (8B aligned).
    const float* xp = x + (size_t)rowbase * DIM + xoff + 2 * h;
    v2f A1 = *(const v2f*)(xp);
    v2f A2 = *(const v2f*)(xp + 4);

    C = __builtin_amdgcn_wmma_f32_16x16x4_f32(false, A1, false, B1,
                                              (short)0, C, false, false);
    C = __builtin_amdgcn_wmma_f32_16x16x4_f32(false, A2, false, B2,
                                              (short)0, C, false, false);
}

__global__ __launch_bounds__(256) void gather_wmma_kernel(
    const float* __restrict__ x,
    const float* __restrict__ w,
    const int*      __restrict__ offsets,
    const EdgeDesc* __restrict__ desc,
    float* __restrict__ out)
{
    const int lane = threadIdx.x & 31;
    const int t    = threadIdx.x >> 5;       // batch tile 0..7 (wave id)
    const int g    = blockIdx.x;             // destination group

    const int h   = lane >> 4;               // K-half / C row-half select
    const int m   = lane & 15;               // A row within tile / B column
    const int col = m & 7;
    const float mask = (m < 8) ? 1.0f : 0.0f;
    const int rowbase = t * 16 + m;

    const int start = __builtin_amdgcn_readfirstlane(offsets[g]);
    const int end   = __builtin_amdgcn_readfirstlane(offsets[g + 1]);
    const int cnt   = end - start;

    // Two ping-pong accumulators break the serial WMMA C-dependency chain.
    v8f Ca = {0.f, 0.f, 0.f, 0.f, 0.f, 0.f, 0.f, 0.f};
    v8f Cb = {0.f, 0.f, 0.f, 0.f, 0.f, 0.f, 0.f, 0.f};

    if (cnt > 0) {
        // Software pipeline with UNCONDITIONAL 2-ahead lookahead (desc is
        // sentinel-padded): loop body is two back-to-back s_load_b64 (one
        // kmcnt wait) + straight-line WMMA work, no inner branches.
        EdgeDesc d0 = desc[start];
        int i = 0;
        for (; i + 1 < cnt; i += 2) {
            EdgeDesc d1 = desc[start + i + 1];
            EdgeDesc d2 = desc[start + i + 2];   // may touch sentinel, never consumed
            edge_accum(x, w, d0, rowbase, h, col, mask, Ca);
            edge_accum(x, w, d1, rowbase, h, col, mask, Cb);
            d0 = d2;
        }
        if (i < cnt) {
            edge_accum(x, w, d0, rowbase, h, col, mask, Ca);
        }
    }
    const v8f C = Ca + Cb;

    // Epilogue: lane (h, col) holds C rows M = r + 8h, column N = col.
    // Fuse the residual: out = x + C. Each output element written exactly once.
    if (m < 8) {
        const size_t colg = (size_t)g * BLK + col;
        #pragma unroll
        for (int r = 0; r < 8; ++r) {
            const size_t idx = (size_t)(t * 16 + 8 * h + r) * DIM + colg;
            out[idx] = x[idx] + C[r];
        }
    }
}

// ---------------- Launch ----------------

extern "C" void kernel_launch(void* const* d_in, const int* in_sizes, int n_in,
                              void* d_out, int out_size, void* d_ws, size_t ws_size,
                              hipStream_t stream)
{
    const float* x       = (const float*)d_in[0];  // (128, 32000)
    const float* w       = (const float*)d_in[1];  // (132000, 8, 8)
    const int* block_in  = (const int*)d_in[2];    // (132000,)
    const int* block_out = (const int*)d_in[3];    // (132000,)
    float* out           = (float*)d_out;          // (128, 32000)

    // Workspace (rebuilt deterministically every call):
    //   count[4096] | offsets[4096] | cursor[4096] | desc[NBLOCKS+2] (~1.1 MB)
    int* count      = (int*)d_ws;
    int* offsets    = count + 4096;
    int* cursor     = offsets + 4096;
    EdgeDesc* desc  = (EdgeDesc*)(cursor + 4096);

    init_kernel<<<(G_GROUPS + 255) / 256, 256, 0, stream>>>(count, desc);
    count_kernel<<<(NBLOCKS + 255) / 256, 256, 0, stream>>>(block_out, count);
    scan_kernel<<<1, SCAN_T, 0, stream>>>(count, offsets, cursor);
    fill_kernel<<<(NBLOCKS + 255) / 256, 256, 0, stream>>>(block_out, block_in,
                                                           cursor, desc);

    // 4000 CTAs x 8 waves: the 8 waves of a CTA share a group's edge list, so
    // w (34 MB) streams from HBM once and re-reads hit L2 (192 MB).
    gather_wmma_kernel<<<G_GROUPS, 256, 0, stream>>>(x, w, offsets, desc, out);
}